// GraphAttentionLayer_29420525978024
// MI455X (gfx1250) — compile-verified
//
#include <hip/hip_runtime.h>
#include <hip/hip_bf16.h>

// Problem constants (match reference setup_inputs)
#define BB   4
#define NN   20000
#define EE   320000
#define INF  128
#define HH   4
#define FF   32
#define OUTF 128   // HH*FF

typedef float v2f __attribute__((ext_vector_type(2)));
typedef float v8f __attribute__((ext_vector_type(8)));

// ---------------------------------------------------------------------------
// Zero a float buffer (grid-stride)
// ---------------------------------------------------------------------------
__global__ void gat_zero(float* __restrict__ p, long long n) {
    long long i = (long long)blockIdx.x * blockDim.x + threadIdx.x;
    long long stride = (long long)gridDim.x * blockDim.x;
    for (; i < n; i += stride) p[i] = 0.0f;
}

// ---------------------------------------------------------------------------
// h = x @ W   (M = BB*NN = 80000 rows, K = 128, N = 128)
// One wave -> one 16x16 tile via V_WMMA_F32_16X16X4_F32, K unrolled 32x.
// Block = 256 threads = 8 waves = one 16-row strip covering all 128 columns.
// ---------------------------------------------------------------------------
__global__ __launch_bounds__(256) void gat_gemm_wmma(const float* __restrict__ x,
                                                     const float* __restrict__ W,
                                                     float* __restrict__ h) {
    __shared__ float As[16][132];           // 16x128 A strip, padded (132%64=4 -> conflict-free)
    const int rowBase = blockIdx.x * 16;
    const int tid = threadIdx.x;

    // Cooperative load: 16*128 floats = 512 float4, 2 per thread
    {
        const float4* src = (const float4*)(x + (size_t)rowBase * INF);
        #pragma unroll
        for (int i = tid; i < 512; i += 256) {
            const int r  = i >> 5;          // 32 float4 per row
            const int c4 = (i & 31) * 4;
            float4 v = src[i];
            As[r][c4 + 0] = v.x;
            As[r][c4 + 1] = v.y;
            As[r][c4 + 2] = v.z;
            As[r][c4 + 3] = v.w;
        }
    }
    __syncthreads();

    const int wave    = tid >> 5;
    const int lane    = tid & 31;
    const int colBase = wave * 16;          // 8 waves x 16 cols = 128
    const int mn      = lane & 15;          // M for A-frag, N for B/C-frags
    const int koff    = (lane >> 4) * 2;    // lanes 16-31 hold K+2/K+3

    v8f c = {};
    #pragma unroll
    for (int k0 = 0; k0 < INF; k0 += 4) {
        v2f a, b;
        a.x = As[mn][k0 + koff];
        a.y = As[mn][k0 + koff + 1];
        const float* Wp = W + (size_t)(k0 + koff) * OUTF + colBase + mn;
        b.x = Wp[0];
        b.y = Wp[OUTF];
        // 8 args: (neg_a, A, neg_b, B, c_mod, C, reuse_a, reuse_b)
        c = __builtin_amdgcn_wmma_f32_16x16x4_f32(false, a, false, b,
                                                  (short)0, c, false, false);
    }

    // C/D layout: VGPR v -> M = v + 8*lane[4], N = lane&15
    float* outp = h + (size_t)(rowBase + (lane >> 4) * 8) * OUTF + colBase + mn;
    #pragma unroll
    for (int v = 0; v < 8; ++v)
        outp[(size_t)v * OUTF] = c[v];
}

// ---------------------------------------------------------------------------
// Per-node attention logits: one wave per (b*N+n, head)
//   alpha_src = sum_f h[node,head,f]*a[head,f]
//   alpha_dst = sum_f h[node,head,f]*a[head,F+f]
// ---------------------------------------------------------------------------
__global__ __launch_bounds__(256) void gat_alpha(const float* __restrict__ h,
                                                 const float* __restrict__ a,
                                                 float* __restrict__ alpha_src,
                                                 float* __restrict__ alpha_dst) {
    const int wid  = (int)((blockIdx.x * blockDim.x + threadIdx.x) >> 5);
    const int lane = threadIdx.x & 31;
    const int total = BB * NN * HH;
    if (wid >= total) return;
    const int head = wid & (HH - 1);
    const int node = wid >> 2;              // flattened b*N+n

    const float hv = h[(size_t)node * OUTF + head * FF + lane];
    float s = hv * a[head * 2 * FF + lane];
    float d = hv * a[head * 2 * FF + FF + lane];
    #pragma unroll
    for (int off = 16; off > 0; off >>= 1) {
        s += __shfl_down(s, off, 32);
        d += __shfl_down(d, off, 32);
    }
    if (lane == 0) {
        alpha_src[wid] = s;
        alpha_dst[wid] = d;
    }
}

// ---------------------------------------------------------------------------
// Edge pass 1: att_sum[b,dst,h] += exp(leaky(alpha_src[src]+alpha_dst[dst]))
// One thread per (b, e, head).
// ---------------------------------------------------------------------------
__global__ void gat_attsum(const long long* __restrict__ ei,
                           const float* __restrict__ alpha_src,
                           const float* __restrict__ alpha_dst,
                           float* __restrict__ att_sum) {
    const long long t = (long long)blockIdx.x * blockDim.x + threadIdx.x;
    const long long total = (long long)BB * EE * HH;
    if (t >= total) return;
    const int head = (int)(t & (HH - 1));
    const long long be = t >> 2;
    const int e = (int)(be % EE);
    const int b = (int)(be / EE);
    const int src = (int)ei[e];
    const int dst = (int)ei[EE + e];

    float att = alpha_src[(size_t)(b * NN + src) * HH + head] +
                alpha_dst[(size_t)(b * NN + dst) * HH + head];
    att = att >= 0.0f ? att : 0.2f * att;
    atomicAdd(&att_sum[(size_t)(b * NN + dst) * HH + head], __expf(att));
}

// ---------------------------------------------------------------------------
// Edge pass 2: out[b,dst,:] += h[b,src,:] * norm(head)
// One wave per (b, e); lane handles 4 consecutive features (float4 gather,
// 4x global_atomic_add_f32 scatter).
// ---------------------------------------------------------------------------
__global__ __launch_bounds__(256) void gat_scatter(const long long* __restrict__ ei,
                                                   const float* __restrict__ h,
                                                   const float* __restrict__ alpha_src,
                                                   const float* __restrict__ alpha_dst,
                                                   const float* __restrict__ att_sum,
                                                   float* __restrict__ out) {
    const long long wid = ((long long)blockIdx.x * blockDim.x + threadIdx.x) >> 5;
    const int lane = threadIdx.x & 31;
    const long long totalWaves = (long long)BB * EE;
    if (wid >= totalWaves) return;
    const int e = (int)(wid % EE);
    const int b = (int)(wid / EE);
    const int src = (int)ei[e];
    const int dst = (int)ei[EE + e];

    const int f0   = lane * 4;              // 4 consecutive features per lane
    const int head = lane >> 3;             // f0/FF : all 4 feats in same head

    float att = alpha_src[(size_t)(b * NN + src) * HH + head] +
                alpha_dst[(size_t)(b * NN + dst) * HH + head];
    att = att >= 0.0f ? att : 0.2f * att;
    const float ex = __expf(att);
    const float norm = ex / (att_sum[(size_t)(b * NN + dst) * HH + head] + 1e-10f);

    const float4 hv = *(const float4*)(h + (size_t)(b * NN + src) * OUTF + f0);
    float* op = out + (size_t)(b * NN + dst) * OUTF + f0;
    atomicAdd(op + 0, hv.x * norm);
    atomicAdd(op + 1, hv.y * norm);
    atomicAdd(op + 2, hv.z * norm);
    atomicAdd(op + 3, hv.w * norm);
}

// ---------------------------------------------------------------------------
extern "C" void kernel_launch(void* const* d_in, const int* in_sizes, int n_in,
                              void* d_out, int out_size, void* d_ws, size_t ws_size,
                              hipStream_t stream) {
    const float*     x  = (const float*)d_in[0];
    const long long* ei = (const long long*)d_in[1];   // int64 edge_index (2,E)
    const float*     W  = (const float*)d_in[2];
    const float*     a  = (const float*)d_in[3];
    float* out = (float*)d_out;

    // Workspace layout (floats)
    float* ws        = (float*)d_ws;
    float* h         = ws;                                   // 80000*128 = 10,240,000
    float* alpha_src = h + (size_t)BB * NN * OUTF;           // 320,000
    float* alpha_dst = alpha_src + (size_t)BB * NN * HH;     // 320,000
    float* att_sum   = alpha_dst + (size_t)BB * NN * HH;     // 320,000

    // 1) zero accumulators
    gat_zero<<<2048, 256, 0, stream>>>(out, (long long)BB * NN * OUTF);
    gat_zero<<<512, 256, 0, stream>>>(att_sum, (long long)BB * NN * HH);

    // 2) h = x @ W   (5000 blocks x 8 waves, one 16x16 WMMA tile per wave)
    gat_gemm_wmma<<<(BB * NN) / 16, 256, 0, stream>>>(x, W, h);

    // 3) per-node alphas: B*N*H = 320000 waves, 8 waves/block
    gat_alpha<<<(BB * NN * HH) / 8, 256, 0, stream>>>(h, a, alpha_src, alpha_dst);

    // 4) edge softmax denominators: B*E*H = 5,120,000 threads
    gat_attsum<<<(BB * EE * HH) / 256, 256, 0, stream>>>(ei, alpha_src, alpha_dst, att_sum);

    // 5) weighted scatter: B*E = 1,280,000 waves, 8 waves/block
    gat_scatter<<<(BB * EE) / 8, 256, 0, stream>>>(ei, h, alpha_src, alpha_dst,
                                                   att_sum, out);
}